// channel_crossmodel_56556129353805
// MI455X (gfx1250) — compile-verified
//
// MI455X (gfx1250): four-step WMMA-f32 FFT pipeline, v3.
// FFT_1024 = F32x32 · twiddle · F32x32 per wave32, all operands in LDS,
// each 32x32 complex matmul = 4 real matmuls done with V_WMMA_F32_16X16X4_F32.
// All DFT/twiddle tables precomputed into LDS (no transcendentals in the hot
// loop); fft routine force-inlined so LDS accesses lower to direct ds_load.
// Deterministic fixed-order reductions everywhere (no float atomics).

#include <hip/hip_runtime.h>
#include <math.h>

constexpr int B_  = 8;
constexpr int C_  = 512;
constexpr int L_  = 1024;
constexpr int KN_ = 4;
constexpr int KS_ = 7;

typedef float v2f __attribute__((ext_vector_type(2)));
typedef float v8f __attribute__((ext_vector_type(8)));

#if defined(__HIP_DEVICE_COMPILE__)
#if !__has_builtin(__builtin_amdgcn_wmma_f32_16x16x4_f32)
#error "device pass: wmma_f32_16x16x4_f32 builtin not available"
#endif
#endif

__device__ __forceinline__ v8f wmma4(v2f a, v2f b, v8f c) {
#if defined(__HIP_DEVICE_COMPILE__)
  // (neg_a, A, neg_b, B, c_mod, C, reuse_a, reuse_b); float types: NEG[0:1]=0
  return __builtin_amdgcn_wmma_f32_16x16x4_f32(false, a, false, b, (short)0, c,
                                               false, false);
#else
  (void)a; (void)b;
  return c;  // host stub; never executed
#endif
}

// Tables: Fc = cos(2*pi*r*c/32), Fsp = +sin, Fsm = -sin (32x32 DFT basis),
// Twc/Tws = cos/sin(2*pi*t/1024) for t = row*col (t < 1024).
__device__ __forceinline__ void initTables(float* Fc, float* Fsp, float* Fsm,
                                           float* Twc, float* Tws, int lane) {
  for (int i = lane; i < 1024; i += 32) {
    int r = i >> 5, c = i & 31;
    float ang = 6.28318530717958647692f * (float)((r * c) & 31) * (1.0f / 32.0f);
    float s, ct;
    __sincosf(ang, &s, &ct);
    Fc[i] = ct; Fsp[i] = s; Fsm[i] = -s;
    float angt = 6.28318530717958647692f * (float)i * (1.0f / 1024.0f);
    float st, ctt;
    __sincosf(angt, &st, &ctt);
    Twc[i] = ctt; Tws[i] = st;
  }
}

// 1024-pt DFT: out[o] = scale * sum_j in[j] * exp(sgn*2*pi*i*j*o/1024).
// Pass Fi = (sgn*sin) table, Fin = (-sgn*sin) table, tsgn = sgn for twiddle.
// Input/output buffers are linear in l.  One wave32; caller syncs before
// (input ready) and after (before reading Or/Oi).
__device__ __forceinline__ void fft1024(
    const float* Fc, const float* Fi_t, const float* Fin_t,
    const float* Twc, const float* Tws, float tsgn,
    const float* Ar, const float* Ai,
    float* Tr, float* Ti, float* Or, float* Oi,
    float scale, int lane) {
  const int n = lane & 15, h = lane >> 4;
  // ---- step 1: T[k1][n2] = sum_n1 F[k1][n1]*A[n1][n2], then twiddle ----
  for (int mi = 0; mi < 2; ++mi)
    for (int ni = 0; ni < 2; ++ni) {
      v8f cr = {0.f, 0.f, 0.f, 0.f, 0.f, 0.f, 0.f, 0.f};
      v8f ci = {0.f, 0.f, 0.f, 0.f, 0.f, 0.f, 0.f, 0.f};
#pragma unroll
      for (int kc = 0; kc < 8; ++kc) {
        const int k0 = kc * 4 + 2 * h;
        const int arow = (mi * 16 + n) * 32 + k0;
        v2f fr, fi, fin, br, bi;
        fr.x = Fc[arow];     fr.y = Fc[arow + 1];
        fi.x = Fi_t[arow];   fi.y = Fi_t[arow + 1];
        fin.x = Fin_t[arow]; fin.y = Fin_t[arow + 1];
        const int bcol = ni * 16 + n;
        br.x = Ar[k0 * 32 + bcol];  br.y = Ar[(k0 + 1) * 32 + bcol];
        bi.x = Ai[k0 * 32 + bcol];  bi.y = Ai[(k0 + 1) * 32 + bcol];
        cr = wmma4(fr, br, cr);   // Fr*Ar
        cr = wmma4(fin, bi, cr);  // -Fi*Ai
        ci = wmma4(fr, bi, ci);   // Fr*Ai
        ci = wmma4(fi, br, ci);   // Fi*Ar
      }
#pragma unroll
      for (int r = 0; r < 8; ++r) {
        int row = mi * 16 + r + 8 * h;  // k1
        int col = ni * 16 + n;          // n2
        int t = row * col;              // < 1024
        float ca = Twc[t];
        float sa = tsgn * Tws[t];
        float xr = cr[r], xi = ci[r];
        Tr[row * 32 + col] = xr * ca - xi * sa;
        Ti[row * 32 + col] = xr * sa + xi * ca;
      }
    }
  __syncthreads();
  // ---- step 2: X[k1][k2] = sum_n2 T[k1][n2]*F[n2][k2]; store transposed ----
  for (int mi = 0; mi < 2; ++mi)
    for (int ni = 0; ni < 2; ++ni) {
      v8f cr = {0.f, 0.f, 0.f, 0.f, 0.f, 0.f, 0.f, 0.f};
      v8f ci = {0.f, 0.f, 0.f, 0.f, 0.f, 0.f, 0.f, 0.f};
#pragma unroll
      for (int kc = 0; kc < 8; ++kc) {
        const int k0 = kc * 4 + 2 * h;
        const int arow = (mi * 16 + n) * 32 + k0;
        v2f tr, ti, tin, fr, fi;
        tr.x = Tr[arow];      tr.y = Tr[arow + 1];
        ti.x = Ti[arow];      ti.y = Ti[arow + 1];
        tin.x = -ti.x;        tin.y = -ti.y;
        const int bcol = ni * 16 + n;
        fr.x = Fc[k0 * 32 + bcol];    fr.y = Fc[(k0 + 1) * 32 + bcol];
        fi.x = Fi_t[k0 * 32 + bcol];  fi.y = Fi_t[(k0 + 1) * 32 + bcol];
        cr = wmma4(tr, fr, cr);
        cr = wmma4(tin, fi, cr);
        ci = wmma4(tr, fi, ci);
        ci = wmma4(ti, fr, ci);
      }
#pragma unroll
      for (int r = 0; r < 8; ++r) {
        int row = mi * 16 + r + 8 * h;  // o1
        int col = ni * 16 + n;          // o2
        Or[row + 32 * col] = cr[r] * scale;  // linear index o1 + 32*o2
        Oi[row + 32 * col] = ci[r] * scale;
      }
    }
}

// ---------------- small prep kernels ----------------

__global__ void k_prep(const float* __restrict__ sig,
                       const float* __restrict__ csh,
                       float* __restrict__ kern) {
  __shared__ float pdf[KN_ * KS_];
  int t = threadIdx.x;
  if (t < KN_ * KS_) {
    int k = t / KS_, j = t % KS_;
    float grid = -3.0f + (float)j;  // linspace(-3,3,7)
    float d = (grid - csh[k]) / (sig[k] + 0.001f);
    pdf[t] = expf(-0.5f * d * d);
  }
  __syncthreads();
  if (t < KN_ * KS_) {
    int k = t / KS_;
    float ssum = 0.01f;
    for (int i = 0; i < KS_; ++i) ssum += pdf[k * KS_ + i];
    kern[t] = pdf[t] / ssum;
  }
}

__global__ void k_chansum(const float* __restrict__ x, float* __restrict__ s) {
  int i = blockIdx.x * blockDim.x + threadIdx.x;  // b*L + l
  if (i >= B_ * L_) return;
  int b = i / L_, l = i % L_;
  const float* xp = x + (size_t)b * C_ * L_ + l;
  float acc = 0.f;
  for (int c = 0; c < C_; ++c) acc += xp[(size_t)c * L_];
  s[i] = acc;
}

__global__ void k_blur(const float* __restrict__ s,
                       const float* __restrict__ kern,
                       float* __restrict__ xs) {
  int i = blockIdx.x * blockDim.x + threadIdx.x;
  if (i >= B_ * KN_ * L_) return;
  int l = i % L_;
  int bk = i / L_;
  int k = bk % KN_, b = bk / KN_;
  const float* sb = s + (size_t)b * L_;
  float acc = 0.f;
#pragma unroll
  for (int t = 0; t < KS_; ++t) {
    int j = l + t - 3;
    j = j < 0 ? 0 : (j > L_ - 1 ? L_ - 1 : j);  // replicate (edge) pad
    acc += sb[j] * kern[k * KS_ + t];
  }
  xs[i] = acc;
}

// ---------------- FFT kernels ----------------

__global__ void __launch_bounds__(32) k_fft_small(
    const float* __restrict__ xs, const float* __restrict__ q,
    float* __restrict__ vfr, float* __restrict__ vfi,
    float* __restrict__ qfr, float* __restrict__ qfi) {
  __shared__ float Fc[1024], Fsp[1024], Fsm[1024], Twc[1024], Tws[1024];
  __shared__ float Ar[1024], Ai[1024], Tr[1024], Ti[1024], Or[1024], Oi[1024];
  int lane = threadIdx.x;
  initTables(Fc, Fsp, Fsm, Twc, Tws, lane);
  const float* in;
  float *orr, *oii;
  if (blockIdx.x < B_ * KN_) {
    in = xs + (size_t)blockIdx.x * L_;
    orr = vfr + (size_t)blockIdx.x * L_;
    oii = vfi + (size_t)blockIdx.x * L_;
  } else {
    int b = blockIdx.x - B_ * KN_;
    in = q + (size_t)b * L_;
    orr = qfr + (size_t)b * L_;
    oii = qfi + (size_t)b * L_;
  }
  for (int i = lane; i < 1024; i += 32) { Ar[i] = in[i]; Ai[i] = 0.f; }
  __syncthreads();
  // forward: F = cos - i*sin -> Fi table = Fsm, Fin table = Fsp, tsgn = -1
  fft1024(Fc, Fsm, Fsp, Twc, Tws, -1.f, Ar, Ai, Tr, Ti, Or, Oi, 1.f, lane);
  __syncthreads();
  for (int i = lane; i < 1024; i += 32) { orr[i] = Or[i]; oii[i] = Oi[i]; }
}

// global complex means of vfs and qf (deterministic fixed-order tree reduce)
__global__ void __launch_bounds__(256) k_means(
    const float* __restrict__ vfr, const float* __restrict__ vfi,
    const float* __restrict__ qfr, const float* __restrict__ qfi,
    float* __restrict__ stats) {
  __shared__ float r0[256], r1[256], r2[256], r3[256];
  int tid = threadIdx.x;
  float svr = 0, svi = 0, sqr = 0, sqi = 0;
  for (int i = tid; i < B_ * KN_ * L_; i += 256) { svr += vfr[i]; svi += vfi[i]; }
  for (int i = tid; i < B_ * L_; i += 256)       { sqr += qfr[i]; sqi += qfi[i]; }
  r0[tid] = svr; r1[tid] = svi; r2[tid] = sqr; r3[tid] = sqi;
  __syncthreads();
  for (int st = 128; st > 0; st >>= 1) {
    if (tid < st) {
      r0[tid] += r0[tid + st]; r1[tid] += r1[tid + st];
      r2[tid] += r2[tid + st]; r3[tid] += r3[tid + st];
    }
    __syncthreads();
  }
  if (tid == 0) {
    stats[0] = r0[0] / (float)(B_ * KN_ * L_);  // mean_x re
    stats[1] = r1[0] / (float)(B_ * KN_ * L_);  // mean_x im
    stats[2] = r2[0] / (float)(B_ * L_);        // mean_y re
    stats[3] = r3[0] / (float)(B_ * L_);        // mean_y im
  }
}

// a[b,k] = sum_l (vfs-mean_x)^2 ; bb[b] = sum_l (qf-mean_x)^2  (ref bug kept)
__global__ void __launch_bounds__(256) k_ab(
    const float* __restrict__ vfr, const float* __restrict__ vfi,
    const float* __restrict__ qfr, const float* __restrict__ qfi,
    float* __restrict__ stats) {
  __shared__ float rr[256], ri[256];
  int tid = threadIdx.x;
  float mxr = stats[0], mxi = stats[1];
  float sr = 0, si = 0;
  if (blockIdx.x < B_ * KN_) {
    const float* vr = vfr + (size_t)blockIdx.x * L_;
    const float* vi = vfi + (size_t)blockIdx.x * L_;
    for (int l = tid; l < L_; l += 256) {
      float a = vr[l] - mxr, b = vi[l] - mxi;
      sr += a * a - b * b;
      si += 2.f * a * b;
    }
  } else {
    int b = blockIdx.x - B_ * KN_;
    const float* qr = qfr + (size_t)b * L_;
    const float* qi = qfi + (size_t)b * L_;
    for (int l = tid; l < L_; l += 256) {
      float a = qr[l] - mxr, c = qi[l] - mxi;  // mean_x on purpose (ref bug)
      sr += a * a - c * c;
      si += 2.f * a * c;
    }
  }
  rr[tid] = sr; ri[tid] = si;
  __syncthreads();
  for (int st = 128; st > 0; st >>= 1) {
    if (tid < st) { rr[tid] += rr[tid + st]; ri[tid] += ri[tid + st]; }
    __syncthreads();
  }
  if (tid == 0) {
    if (blockIdx.x < B_ * KN_) {
      stats[4 + 2 * blockIdx.x] = rr[0];
      stats[5 + 2 * blockIdx.x] = ri[0];
    } else {
      int b = blockIdx.x - B_ * KN_;
      stats[68 + 2 * b] = rr[0];
      stats[69 + 2 * b] = ri[0];
    }
  }
}

__global__ void k_weight(const float* __restrict__ vfr,
                         const float* __restrict__ vfi,
                         const float* __restrict__ qfr,
                         const float* __restrict__ qfi,
                         const float* __restrict__ stats,
                         float* __restrict__ w) {
  int i = blockIdx.x * blockDim.x + threadIdx.x;
  if (i >= B_ * KN_ * L_) return;
  int l = i & (L_ - 1);
  int bk = i >> 10;
  int b = bk >> 2;  // KN_=4
  float mxr = stats[0], mxi = stats[1], myr = stats[2], myi = stats[3];
  float ar = stats[4 + 2 * bk], ai = stats[5 + 2 * bk];
  float br_ = stats[68 + 2 * b], bi_ = stats[69 + 2 * b];
  float xcr = vfr[i] - mxr, xci = vfi[i] - mxi;
  float ycr = qfr[(size_t)b * L_ + l] - myr, yci = qfi[(size_t)b * L_ + l] - myi;
  float covr = xcr * ycr - xci * yci;
  float covi = xcr * yci + xci * ycr;
  float denr = ar * br_ - ai * bi_ + 0.001f;
  float deni = ar * bi_ + ai * br_;
  float d2 = denr * denr + deni * deni;
  float rrho = (covr * denr + covi * deni) / d2;
  float irho = (covi * denr - covr * deni) / d2;
  float mag = sqrtf(rrho * rrho + irho * irho);
  w[i] = mag * tanhf(0.5f * atan2f(irho, rrho + 1e-4f));
}

// u[b,k,:] = ifft(qf[b,:] * (1 - w[b,k,:]))
__global__ void __launch_bounds__(32) k_u(
    const float* __restrict__ qfr, const float* __restrict__ qfi,
    const float* __restrict__ w, float* __restrict__ ur, float* __restrict__ ui) {
  __shared__ float Fc[1024], Fsp[1024], Fsm[1024], Twc[1024], Tws[1024];
  __shared__ float Ar[1024], Ai[1024], Tr[1024], Ti[1024], Or[1024], Oi[1024];
  int lane = threadIdx.x;
  int bk = blockIdx.x;
  int b = bk >> 2;
  initTables(Fc, Fsp, Fsm, Twc, Tws, lane);
  for (int i = lane; i < 1024; i += 32) {
    float g = 1.f - w[(size_t)bk * L_ + i];
    Ar[i] = qfr[(size_t)b * L_ + i] * g;
    Ai[i] = qfi[(size_t)b * L_ + i] * g;
  }
  __syncthreads();
  // inverse: F = cos + i*sin -> Fi = Fsp, Fin = Fsm, tsgn = +1
  fft1024(Fc, Fsp, Fsm, Twc, Tws, +1.f, Ar, Ai, Tr, Ti, Or, Oi,
          1.f / 1024.f, lane);
  __syncthreads();
  for (int i = lane; i < 1024; i += 32) {
    ur[(size_t)bk * L_ + i] = Or[i];
    ui[(size_t)bk * L_ + i] = Oi[i];
  }
}

// Main pass: per (b,c): fwd FFT(x) -> xb; for each k: ifft(xb*w) -> y;
// outputs = |u + y| + x.
__global__ void __launch_bounds__(32) k_main(
    const float* __restrict__ x, const float* __restrict__ w,
    const float* __restrict__ ur, const float* __restrict__ ui,
    float* __restrict__ outputs) {
  __shared__ float Fc[1024], Fsp[1024], Fsm[1024], Twc[1024], Tws[1024];
  __shared__ float Ar[1024], Ai[1024], Tr[1024], Ti[1024], Or[1024], Oi[1024];
  __shared__ float Xr[1024], Xi[1024], Xin[1024];
  int lane = threadIdx.x;
  int bc = blockIdx.x;
  int b = bc / C_;
  initTables(Fc, Fsp, Fsm, Twc, Tws, lane);
  const float* xin = x + (size_t)bc * L_;
  for (int i = lane; i < 1024; i += 32) {
    float v = xin[i];
    Xin[i] = v; Ar[i] = v; Ai[i] = 0.f;
  }
  __syncthreads();
  fft1024(Fc, Fsm, Fsp, Twc, Tws, -1.f, Ar, Ai, Tr, Ti, Xr, Xi, 1.f, lane);
  __syncthreads();
  for (int k = 0; k < KN_; ++k) {
    const float* wk = w + ((size_t)(b * KN_ + k)) * L_;
    for (int i = lane; i < 1024; i += 32) {
      float wv = wk[i];
      Ar[i] = Xr[i] * wv;
      Ai[i] = Xi[i] * wv;
    }
    __syncthreads();
    fft1024(Fc, Fsp, Fsm, Twc, Tws, +1.f, Ar, Ai, Tr, Ti, Or, Oi,
            1.f / 1024.f, lane);
    __syncthreads();
    const float* urk = ur + ((size_t)(b * KN_ + k)) * L_;
    const float* uik = ui + ((size_t)(b * KN_ + k)) * L_;
    float* outk = outputs + (((size_t)bc * KN_) + k) * L_;
    for (int i = lane; i < 1024; i += 32) {
      float yr = Or[i] + urk[i], yi = Oi[i] + uik[i];
      outk[i] = sqrtf(yr * yr + yi * yi) + Xin[i];
    }
    __syncthreads();
  }
}

// BN stats straight from `outputs` in fixed order (deterministic).
__global__ void k_bnstats(const float* __restrict__ outputs,
                          float* __restrict__ mu, float* __restrict__ rstd) {
  int l = blockIdx.x * blockDim.x + threadIdx.x;
  if (l >= L_) return;
  float s = 0.f, sq = 0.f;
  for (int j = 0; j < B_ * C_ * KN_; ++j) {  // fixed order
    float o = outputs[(size_t)j * L_ + l];
    s += o;
    sq += o * o;
  }
  float n = (float)(B_ * C_ * KN_);
  float m = s / n;
  float v = sq / n - m * m;  // biased var (jnp.var default)
  mu[l] = m;
  rstd[l] = rsqrtf(v + 1e-5f);
}

__global__ void __launch_bounds__(256) k_final(
    const float* __restrict__ outputs, const float* __restrict__ q,
    const float* __restrict__ mu, const float* __restrict__ rstd,
    const float* __restrict__ gamma, const float* __restrict__ beta,
    float* __restrict__ out) {
  __shared__ float on[KN_][L_];
  __shared__ float red[256];
  __shared__ float sco[KN_];
  __shared__ float wk[KN_];
  int bc = blockIdx.x;
  int b = bc / C_;
  int tid = threadIdx.x;
  const float* qb = q + (size_t)b * L_;
  for (int k = 0; k < KN_; ++k)
    for (int i = tid; i < L_; i += 256) {
      float o = outputs[(((size_t)bc * KN_) + k) * L_ + i];
      on[k][i] = (o - mu[i]) * rstd[i] * gamma[i] + beta[i];
    }
  __syncthreads();
  for (int k = 0; k < KN_; ++k) {
    float acc = 0.f;
    for (int i = tid; i < L_; i += 256) acc += qb[i] * on[k][i];
    red[tid] = acc;
    __syncthreads();
    for (int st = 128; st > 0; st >>= 1) {
      if (tid < st) red[tid] += red[tid + st];
      __syncthreads();
    }
    if (tid == 0) sco[k] = red[0];
    __syncthreads();
  }
  if (tid == 0) {
    float mx = sco[0];
    for (int k = 1; k < KN_; ++k) mx = fmaxf(mx, sco[k]);
    float s = 0.f;
    for (int k = 0; k < KN_; ++k) { wk[k] = expf(sco[k] - mx); s += wk[k]; }
    for (int k = 0; k < KN_; ++k) wk[k] /= s;
  }
  __syncthreads();
  float* ob = out + (size_t)bc * L_;
  for (int i = tid; i < L_; i += 256) {
    float acc = 0.f;
    for (int k = 0; k < KN_; ++k) acc += wk[k] * on[k][i];
    ob[i] = acc;
  }
  if (bc == 0 && tid == 0) out[(size_t)B_ * C_ * L_] = 0.0f;  // tuple's 0.0
}

// ---------------- launch ----------------

extern "C" void kernel_launch(void* const* d_in, const int* in_sizes, int n_in,
                              void* d_out, int out_size, void* d_ws,
                              size_t ws_size, hipStream_t stream) {
  (void)in_sizes; (void)n_in; (void)out_size; (void)ws_size;
  const float* x     = (const float*)d_in[0];
  const float* q     = (const float*)d_in[1];
  const float* sig   = (const float*)d_in[2];
  const float* csh   = (const float*)d_in[3];
  const float* gamma = (const float*)d_in[4];
  const float* beta  = (const float*)d_in[5];
  float* out = (float*)d_out;
  float* ws = (float*)d_ws;

  size_t o = 0;
  auto alloc = [&](size_t n) { float* p = ws + o; o += n; return p; };
  float* kern = alloc(32);
  float* s    = alloc((size_t)B_ * L_);
  float* xs   = alloc((size_t)B_ * KN_ * L_);
  float* w    = alloc((size_t)B_ * KN_ * L_);
  float* vfr  = alloc((size_t)B_ * KN_ * L_);
  float* vfi  = alloc((size_t)B_ * KN_ * L_);
  float* qfr  = alloc((size_t)B_ * L_);
  float* qfi  = alloc((size_t)B_ * L_);
  float* ur   = alloc((size_t)B_ * KN_ * L_);
  float* ui   = alloc((size_t)B_ * KN_ * L_);
  float* stats= alloc(128);
  float* mu   = alloc(L_);
  float* rstd = alloc(L_);
  float* outputs = alloc((size_t)B_ * C_ * KN_ * L_);  // 64 MB

  k_prep<<<1, 32, 0, stream>>>(sig, csh, kern);
  k_chansum<<<(B_ * L_ + 255) / 256, 256, 0, stream>>>(x, s);
  k_blur<<<(B_ * KN_ * L_ + 255) / 256, 256, 0, stream>>>(s, kern, xs);
  k_fft_small<<<B_ * KN_ + B_, 32, 0, stream>>>(xs, q, vfr, vfi, qfr, qfi);
  k_means<<<1, 256, 0, stream>>>(vfr, vfi, qfr, qfi, stats);
  k_ab<<<B_ * KN_ + B_, 256, 0, stream>>>(vfr, vfi, qfr, qfi, stats);
  k_weight<<<(B_ * KN_ * L_ + 255) / 256, 256, 0, stream>>>(vfr, vfi, qfr, qfi,
                                                            stats, w);
  k_u<<<B_ * KN_, 32, 0, stream>>>(qfr, qfi, w, ur, ui);
  k_main<<<B_ * C_, 32, 0, stream>>>(x, w, ur, ui, outputs);
  k_bnstats<<<(L_ + 255) / 256, 256, 0, stream>>>(outputs, mu, rstd);
  k_final<<<B_ * C_, 256, 0, stream>>>(outputs, q, mu, rstd, gamma, beta, out);
}